// ValueQuantizer_68616397520952
// MI455X (gfx1250) — compile-verified
//
#include <hip/hip_runtime.h>
#include <hip/hip_fp16.h>
#include <stdint.h>

// ---------------------------------------------------------------------------
// 4-bit group quantizer (GROUP=32, d=128) for MI455X / gfx1250.
// Memory-bound streaming kernel: async global->LDS double-buffered staging
// (ASYNCcnt path), wave32 shuffle reductions, non-temporal output stores.
// ---------------------------------------------------------------------------

#define D        128          // row length (last dim)
#define WPB      8            // waves per block (wave32)
#define TPB      (WPB * 32)   // 256 threads

typedef float v4f __attribute__((ext_vector_type(4)));
typedef float v2f __attribute__((ext_vector_type(2)));

// Issue one wave-wide async copy: 32 lanes x 16B = one 512B row -> LDS.
// VDST = per-lane LDS byte address, VADDR = per-lane 64-bit global address.
__device__ __forceinline__ void async_row_to_lds(unsigned lds_byte_addr,
                                                 const float* gptr) {
    unsigned long long ga = (unsigned long long)(uintptr_t)gptr;
    asm volatile("global_load_async_to_lds_b128 %0, %1, off"
                 :: "v"(lds_byte_addr), "v"(ga)
                 : "memory");
}

__global__ __launch_bounds__(TPB) void ValueQuantizer_gfx1250_kernel(
    const float* __restrict__ v,
    float*       __restrict__ out_packed,   // rows * 64
    float*       __restrict__ out_scales,   // rows * 4
    float*       __restrict__ out_zeros,    // rows * 4
    float*       __restrict__ out_vhat,     // rows * 128
    int rows)
{
    // Two 128-float (512B) row buffers per wave: 8 KiB of the 320 KiB WGP LDS.
    __shared__ float smem[WPB * 2 * D];

    const int lane        = threadIdx.x & 31;
    const int wave        = threadIdx.x >> 5;
    const int wave_global = blockIdx.x * WPB + wave;
    const int wave_stride = gridDim.x * WPB;

    float* mybuf = &smem[wave * (2 * D)];
    // Flat->LDS address: low 32 bits of the flat shared pointer are the LDS
    // byte offset (aperture lives in the high 32 bits on gfx1250).
    const unsigned lds_base = (unsigned)(uintptr_t)mybuf;
    const unsigned lds0 = lds_base + (unsigned)(lane * 16);
    const unsigned lds1 = lds0 + (unsigned)(D * 4);

    int row = wave_global;
    if (row < rows)
        async_row_to_lds(lds0, v + (size_t)row * D + lane * 4);   // prologue

    int buf = 0;
    for (; row < rows; row += wave_stride) {
        const int nxt = row + wave_stride;
        if (nxt < rows) {
            // Prefetch next row into the other buffer, then wait for the
            // OLDER of the two in-flight async copies (in-order retirement).
            async_row_to_lds(buf ? lds0 : lds1,
                             v + (size_t)nxt * D + lane * 4);
            asm volatile("s_wait_asynccnt 0x1" ::: "memory");
        } else {
            asm volatile("s_wait_asynccnt 0x0" ::: "memory");
        }

        // Lane L holds elements [4L, 4L+4) of the row (ds_load_b128).
        v4f x = *(const v4f*)(mybuf + buf * D + lane * 4);

        // Group = 32 elements = 8 lanes; butterfly min/max inside 8-lane sets.
        float mn = fminf(fminf(x.x, x.y), fminf(x.z, x.w));
        float mx = fmaxf(fmaxf(x.x, x.y), fmaxf(x.z, x.w));
        #pragma unroll
        for (int m = 1; m <= 4; m <<= 1) {
            mn = fminf(mn, __shfl_xor(mn, m, 32));
            mx = fmaxf(mx, __shfl_xor(mx, m, 32));
        }

        const float scale = fmaxf((mx - mn) / 15.0f, 1e-8f);
        const float zero  = mn;
        const float inv   = 1.0f / scale;
        // f16 round-trip (reference stores f16 scale/zero and dequantizes
        // with the rounded values).
        const float s16 = __half2float(__float2half_rn(scale));
        const float z16 = __half2float(__float2half_rn(zero));

        // RNE rounding (jnp.round), clamp to [0,15].
        const int q0 = (int)fminf(fmaxf(rintf((x.x - zero) * inv), 0.f), 15.f);
        const int q1 = (int)fminf(fmaxf(rintf((x.y - zero) * inv), 0.f), 15.f);
        const int q2 = (int)fminf(fmaxf(rintf((x.z - zero) * inv), 0.f), 15.f);
        const int q3 = (int)fminf(fmaxf(rintf((x.w - zero) * inv), 0.f), 15.f);

        // packed[2L]   = q(4L)   | q(4L+1)<<4
        // packed[2L+1] = q(4L+2) | q(4L+3)<<4
        v2f pk;
        pk.x = (float)(q0 | (q1 << 4));
        pk.y = (float)(q2 | (q3 << 4));
        __builtin_nontemporal_store(
            pk, (v2f*)(out_packed + (size_t)row * (D / 2) + lane * 2));

        // v_hat = idx * s16 + z16  (uses f16-rounded scale/zero)
        v4f vh;
        vh.x = (float)q0 * s16 + z16;
        vh.y = (float)q1 * s16 + z16;
        vh.z = (float)q2 * s16 + z16;
        vh.w = (float)q3 * s16 + z16;
        __builtin_nontemporal_store(
            vh, (v4f*)(out_vhat + (size_t)row * D + lane * 4));

        if ((lane & 7) == 0) {
            const int g = lane >> 3;       // group id 0..3
            __builtin_nontemporal_store(s16, out_scales + (size_t)row * 4 + g);
            __builtin_nontemporal_store(z16, out_zeros  + (size_t)row * 4 + g);
        }

        buf ^= 1;
    }
}

extern "C" void kernel_launch(void* const* d_in, const int* in_sizes, int n_in,
                              void* d_out, int out_size, void* d_ws, size_t ws_size,
                              hipStream_t stream) {
    (void)n_in; (void)out_size; (void)d_ws; (void)ws_size;

    const float* v = (const float*)d_in[0];
    const long long n    = (long long)in_sizes[0];   // 4*32*4096*128
    const int       rows = (int)(n / D);             // 524288

    // Outputs concatenated flat in reference return order, as floats:
    // packed (rows*64) | scales (rows*4) | zeros (rows*4) | v_hat (rows*128)
    float* out        = (float*)d_out;
    float* out_packed = out;
    float* out_scales = out_packed + (size_t)rows * (D / 2);
    float* out_zeros  = out_scales + (size_t)rows * 4;
    float* out_vhat   = out_zeros  + (size_t)rows * 4;

    // 2048 blocks x 8 waves = 16384 waves -> 32 rows per wave (grid-stride).
    dim3 grid(2048), block(TPB);
    hipLaunchKernelGGL(ValueQuantizer_gfx1250_kernel, grid, block, 0, stream,
                       v, out_packed, out_scales, out_zeros, out_vhat, rows);
}